// DotInteraction_42185168781611
// MI455X (gfx1250) — compile-verified
//
#include <hip/hip_runtime.h>

typedef float v2f __attribute__((ext_vector_type(2)));
typedef float v8f __attribute__((ext_vector_type(8)));
typedef int   v4i_vs __attribute__((vector_size(16)));
typedef __attribute__((address_space(1))) v4i_vs* as1_v4i_p;   // global
typedef __attribute__((address_space(3))) v4i_vs* as3_v4i_p;   // LDS

#define FDIM 64
#define DDIM 256
#define LDS_STRIDE 260   // 256 + 4 pad: bank-conflict-free ds_load_b64, 16B-aligned rows
#define NPAIR 2016       // 64*63/2
#define NTHREADS 320     // 10 waves, one 16x16 upper-tri tile each

__global__ __launch_bounds__(NTHREADS)
void DotInteraction_gram_triu_kernel(const float* __restrict__ X,
                                     float* __restrict__ out)
{
    __shared__ float lds[FDIM * LDS_STRIDE];

    const int b = blockIdx.x;
    const int t = threadIdx.x;

    // ---- Stage the 64x256 fp32 tile into padded LDS ----
#if __has_builtin(__builtin_amdgcn_global_load_async_to_lds_b128)
    {
        const float* gbase = X + (size_t)b * FDIM * DDIM;
        #pragma unroll
        for (int it = 0; it < 13; ++it) {
            int g = t + it * NTHREADS;            // float4 index, 4096 total
            if (g < (FDIM * DDIM / 4)) {
                int row = g >> 6;                 // 64 float4 per 256-float row
                int c4  = (g & 63) << 2;
                __builtin_amdgcn_global_load_async_to_lds_b128(
                    (as1_v4i_p)(gbase + g * 4),
                    (as3_v4i_p)&lds[row * LDS_STRIDE + c4],
                    /*offset=*/0, /*cpol=*/0);
            }
        }
        __builtin_amdgcn_s_wait_asynccnt(0);
    }
#else
    {
        const float4* src = reinterpret_cast<const float4*>(X + (size_t)b * FDIM * DDIM);
        #pragma unroll
        for (int it = 0; it < 13; ++it) {
            int g = t + it * NTHREADS;
            if (g < (FDIM * DDIM / 4)) {
                float4 v = src[g];
                int row = g >> 6;
                int c4  = (g & 63) << 2;
                *reinterpret_cast<float4*>(&lds[row * LDS_STRIDE + c4]) = v;
            }
        }
    }
#endif
    __syncthreads();

    // ---- Tile assignment: wave -> (ti, tj), ti <= tj ----
    const int wave = t >> 5;
    const int lane = t & 31;
    int ti, tj;
    if      (wave < 4) { ti = 0; tj = wave;     }
    else if (wave < 7) { ti = 1; tj = wave - 3; }
    else if (wave < 9) { ti = 2; tj = wave - 5; }
    else               { ti = 3; tj = 3;        }

    const int halfsel = lane >> 4;   // 0: lanes 0-15, 1: lanes 16-31
    const int lrow    = lane & 15;
    const int kb      = halfsel * 2; // K sub-offset within a K=4 slice

    // A operand: rows of tile ti; B operand: (transposed) rows of tile tj.
    const v2f* pa = reinterpret_cast<const v2f*>(&lds[(ti * 16 + lrow) * LDS_STRIDE + kb]);
    const v2f* pb = reinterpret_cast<const v2f*>(&lds[(tj * 16 + lrow) * LDS_STRIDE + kb]);

    v8f acc = {0.f, 0.f, 0.f, 0.f, 0.f, 0.f, 0.f, 0.f};

    if (ti == tj) {
        // Diagonal tile: per-lane A and B operand values are identical
        // (both = X[ti*16 + lane%16][k + 2*(lane/16)]) -> load once, halve LDS traffic.
        #pragma unroll 8
        for (int kk = 0; kk < DDIM / 4; ++kk) {
            v2f a = pa[kk * 2];    // byte offset kk*16
            acc = __builtin_amdgcn_wmma_f32_16x16x4_f32(
                false, a, false, a, (short)0, acc, false, false);
        }
    } else {
        #pragma unroll 8
        for (int kk = 0; kk < DDIM / 4; ++kk) {
            v2f a  = pa[kk * 2];
            v2f bb = pb[kk * 2];
            acc = __builtin_amdgcn_wmma_f32_16x16x4_f32(
                false, a, false, bb, (short)0, acc, false, false);
        }
    }

    // ---- Scatter strictly-upper-triangular entries ----
    // C/D layout: VGPR r, lane l -> M = r + 8*(l>=16), N = l&15
    float* o = out + (size_t)b * NPAIR;
    const int N = lrow;
    #pragma unroll
    for (int r = 0; r < 8; ++r) {
        int M = r + (halfsel << 3);
        int i = ti * 16 + M;
        int j = tj * 16 + N;
        if (j > i) {
            int off = i * 63 - (i * (i - 1)) / 2 + (j - i - 1);
            o[off] = acc[r];
        }
    }
}

extern "C" void kernel_launch(void* const* d_in, const int* in_sizes, int n_in,
                              void* d_out, int out_size, void* d_ws, size_t ws_size,
                              hipStream_t stream) {
    const float* X = (const float*)d_in[0];
    float* out = (float*)d_out;
    int B = in_sizes[0] / (FDIM * DDIM);   // 8192
    DotInteraction_gram_triu_kernel<<<B, NTHREADS, 0, stream>>>(X, out);
}